// KFLoss_82600811036958
// MI455X (gfx1250) — compile-verified
//
#include <hip/hip_runtime.h>
#include <math.h>
#include <stdint.h>

#define EPS_KF 1e-6f
#define BLK 256   // threads per block (8 wave32s)
#define TI  16    // pred rows per block
#define JPT 4     // target cols per thread (float4 store)

// ---- CDNA5 feature probes (device pass only; host pass sees fallbacks) ----
#if defined(__HIP_DEVICE_COMPILE__)
  #if __has_builtin(__builtin_amdgcn_global_load_async_to_lds_b128)
    #define HAVE_ASYNC_COPY 1
  #endif
  #if __has_builtin(__builtin_amdgcn_s_wait_asynccnt)
    #define HAVE_WAIT_ASYNC 1
  #endif
#endif

// exact parameter types for the async-copy builtin:
//   arg0: v4i* in AS1 (global), arg1: v4i* in AS3 (LDS)
typedef int v4i __attribute__((ext_vector_type(4)));
typedef __attribute__((address_space(1))) v4i gv4i;
typedef __attribute__((address_space(3))) v4i lv4i;

__device__ __forceinline__ float rcp_fast(float x)  { return __builtin_amdgcn_rcpf(x); }
__device__ __forceinline__ float sqrt_fast(float x) { return __builtin_amdgcn_sqrtf(x); }

__device__ __forceinline__ void wait_async0() {
#if defined(HAVE_WAIT_ASYNC)
  __builtin_amdgcn_s_wait_asynccnt(0);
#elif defined(HAVE_ASYNC_COPY)
  asm volatile("s_wait_asynccnt 0" ::: "memory");
#endif
}

// ---------- per-row sigma components: (s00, s01, s11, Vb) ----------
__global__ void sigma_kernel(const float* __restrict__ pred, int np,
                             const float* __restrict__ tgt,  int nt,
                             float* __restrict__ ppar, float* __restrict__ tpar) {
  int idx = blockIdx.x * blockDim.x + threadIdx.x;
  if (idx >= np + nt) return;
  const float* src;
  float* dst;
  if (idx < np) { src = pred + (size_t)idx * 5; dst = ppar + (size_t)idx * 4; }
  else          { int j = idx - np; src = tgt + (size_t)j * 5; dst = tpar + (size_t)j * 4; }

  float w = fminf(fmaxf(src[2], 1e-7f), 1e7f);
  float h = fminf(fmaxf(src[3], 1e-7f), 1e7f);
  float r = src[4];
  float c = __cosf(r), s = __sinf(r);
  float a = 0.25f * w * w;          // (w/2)^2
  float b = 0.25f * h * h;          // (h/2)^2
  float s00 = a * c * c + b * s * s;
  float s01 = (a - b) * s * c;
  float s11 = a * s * s + b * c * c;
  float det = s00 * s11 - s01 * s01;
  float vb  = 4.0f * sqrt_fast(fabsf(det));
  dst[0] = s00; dst[1] = s01; dst[2] = s11; dst[3] = vb;
}

// ---------- pairwise KFIoU loss for one (p, t) pair ----------
__device__ __forceinline__ float kf_pair(float p00, float p01, float p11, float vbp,
                                         float t00, float t01, float t11, float vbt) {
  float m00 = p00 + t00, m01 = p01 + t01, m11 = p11 + t11;
  float det_m = m00 * m11 - m01 * m01;
  float inv = rcp_fast(det_m);                       // trans #1
  float i00 = m11 * inv, i01 = -m01 * inv, i11 = m00 * inv;
  float k00 = p00 * i00 + p01 * i01;
  float k01 = p00 * i01 + p01 * i11;
  float k10 = p01 * i00 + p11 * i01;
  float k11 = p01 * i01 + p11 * i11;
  float s00 = p00 - (k00 * p00 + k01 * p01);
  float s01 = p01 - (k00 * p01 + k01 * p11);
  float s10 = p01 - (k10 * p00 + k11 * p01);
  float s11 = p11 - (k10 * p01 + k11 * p11);
  float det_s = s00 * s11 - s01 * s10;
  float vb = 4.0f * sqrt_fast(fabsf(det_s));         // trans #2
  float kf = vb * rcp_fast(vbp + vbt - vb + EPS_KF); // trans #3
  float loss = 1.0f - kf;
  return loss > 0.0f ? loss : 0.0f;                  // clip(., 0)
}

// ---------- main pairwise kernel: TI pred rows x (BLK*JPT) target cols ----------
__global__ __launch_bounds__(BLK) void kfiou_kernel(const float* __restrict__ ppar,
                                                    const float* __restrict__ tpar,
                                                    float* __restrict__ out,
                                                    int np, int nt) {
  __shared__ float sp[TI * 4];

  const int tid  = threadIdx.x;
  const int i0   = blockIdx.y * TI;
  const int j0   = (blockIdx.x * BLK + tid) * JPT;
  const int rows = (np - i0) < TI ? (np - i0) : TI;

  // ---- stage pred params for this block's rows into LDS (CDNA5 async copy) ----
#if defined(HAVE_ASYNC_COPY)
  if (tid < rows) {
    const float* g = ppar + (size_t)(i0 + tid) * 4;
    __builtin_amdgcn_global_load_async_to_lds_b128(
        (gv4i*)(uintptr_t)g,                 // AS1 pointer (bit-identical to generic)
        (lv4i*)(uintptr_t)&sp[tid * 4],      // AS3 pointer = low 32 bits of generic
        0, 0);
  }
  wait_async0();
#else
  if (tid < rows) {
    ((float4*)sp)[tid] = ((const float4*)ppar)[i0 + tid];
  }
#endif
  __syncthreads();

  // ---- this thread's JPT target-param quads (held in registers) ----
  int jcnt = (j0 + JPT <= nt) ? JPT : (j0 < nt ? nt - j0 : 0);
  float4 tq[JPT];
  #pragma unroll
  for (int q = 0; q < JPT; ++q)
    if (q < jcnt) tq[q] = ((const float4*)tpar)[j0 + q];

  if (jcnt == 0) return;
  const bool full = (jcnt == JPT) && ((nt & 3) == 0);

  for (int ii = 0; ii < rows; ++ii) {
    // uniform-address LDS read -> broadcast ds_load
    float p00 = sp[ii * 4 + 0];
    float p01 = sp[ii * 4 + 1];
    float p11 = sp[ii * 4 + 2];
    float vbp = sp[ii * 4 + 3];

    float res[JPT];
    #pragma unroll
    for (int q = 0; q < JPT; ++q) {
      if (q < jcnt)
        res[q] = kf_pair(p00, p01, p11, vbp, tq[q].x, tq[q].y, tq[q].z, tq[q].w);
    }

    float* row = out + (size_t)(i0 + ii) * nt + j0;
    if (full) {
      *(float4*)row = make_float4(res[0], res[1], res[2], res[3]);
    } else {
      for (int q = 0; q < jcnt; ++q) row[q] = res[q];
    }
  }
}

extern "C" void kernel_launch(void* const* d_in, const int* in_sizes, int n_in,
                              void* d_out, int out_size, void* d_ws, size_t ws_size,
                              hipStream_t stream) {
  const float* pred = (const float*)d_in[0];
  const float* tgt  = (const float*)d_in[1];
  const int np = in_sizes[0] / 5;   // 4096
  const int nt = in_sizes[1] / 5;   // 2048

  float* ppar = (float*)d_ws;                 // np * 4 floats
  float* tpar = ppar + (size_t)np * 4;        // nt * 4 floats  (total 96 KB)

  const int ntot = np + nt;
  sigma_kernel<<<(ntot + 255) / 256, 256, 0, stream>>>(pred, np, tgt, nt, ppar, tpar);

  dim3 grid((nt + BLK * JPT - 1) / (BLK * JPT), (np + TI - 1) / TI);
  kfiou_kernel<<<grid, BLK, 0, stream>>>(ppar, tpar, (float*)d_out, np, nt);
}